// Generator_81449759801580
// MI455X (gfx1250) — compile-verified
//
#include <hip/hip_runtime.h>
#include <math.h>

// Problem constants from the reference: img (B,C,H,W) = (64,3,512,512) f32.
#define BB 64
#define CC 3
#define HH 512
#define WW 512

// One block = one (batch, row-pair). 256 threads (8 wave32 waves).
// Thread t: row_in_block = t>>7, handles 4 consecutive x pixels at (t&127)*4.
// Per channel it emits one float4 (global_store_b128) -> fully coalesced
// stores across the wave. Gathers for all 3 channels reuse the same 4
// corner offsets/weights (computed once per pixel).
//
// Roofline: ~403 MB HBM traffic vs ~0.7 GFLOP -> pure memory-bound gather;
// each 3 MB batch image is L2-resident (192 MB L2), so corner gathers hit L2.
__global__ __launch_bounds__(256) void warp_bilinear_kernel(
    const float* __restrict__ img,    // (B,C,H,W)
    const float* __restrict__ rot,    // (B,)
    const float* __restrict__ trans,  // (B,2)
    const float* __restrict__ scale,  // (B,)
    float* __restrict__ out)          // (B,C,H,W)
{
    const int tid = threadIdx.x;
    const int b   = blockIdx.y;
    const int y   = blockIdx.x * 2 + (tid >> 7);
    const int xb  = (tid & 127) * 4;

    // Per-batch parameters (uniform within the block -> scalar loads).
    const float r  = rot[b];
    const float sc = scale[b];
    float sn, cs;
    __sincosf(r, &sn, &cs);
    // v_rcp_f32 instead of the precise-division sequence (~30 VALU ops saved);
    // 1-ulp reciprocal is more than adequate here (reference scale == 1.0).
    const float inv = __builtin_amdgcn_rcpf(sc);
    cs *= inv;
    sn *= inv;
    const float tx = trans[2 * b + 0];
    const float ty = trans[2 * b + 1];

    // Source coords of the first pixel of this thread's 4-pixel group.
    // gx = (2x+1)/W - 1 ; gy = (2y+1)/H - 1
    // ix = ((cs*gx - sn*gy + tx + 1)*W - 1)*0.5 ; d(ix)/dx = cs
    // iy = ((sn*gx + cs*gy + ty + 1)*H - 1)*0.5 ; d(iy)/dx = sn
    const float gy  = (2.0f * (float)y  + 1.0f) * (1.0f / (float)HH) - 1.0f;
    const float gx0 = (2.0f * (float)xb + 1.0f) * (1.0f / (float)WW) - 1.0f;
    float ix = ((cs * gx0 - sn * gy + tx + 1.0f) * (float)WW - 1.0f) * 0.5f;
    float iy = ((sn * gx0 + cs * gy + ty + 1.0f) * (float)HH - 1.0f) * 0.5f;

    // Per-pixel corner weights (validity folded in) and clamped in-channel
    // offsets, shared across the 3 channels.
    float w00[4], w01[4], w10[4], w11[4];
    int   o00[4], o01[4], o10[4], o11[4];

#pragma unroll
    for (int j = 0; j < 4; ++j) {
        const float fx = ix + (float)j * cs;
        const float fy = iy + (float)j * sn;
        const float x0f = floorf(fx);
        const float y0f = floorf(fy);
        const float wx1 = fx - x0f;          // weight toward x0+1
        const float wy1 = fy - y0f;          // weight toward y0+1
        const float wx0 = 1.0f - wx1;
        const float wy0 = 1.0f - wy1;

        const int x0 = (int)x0f;
        const int y0 = (int)y0f;
        const int x1 = x0 + 1;
        const int y1 = y0 + 1;

        const float vx0 = (x0 >= 0 && x0 < WW) ? 1.0f : 0.0f;
        const float vx1 = (x1 >= 0 && x1 < WW) ? 1.0f : 0.0f;
        const float vy0 = (y0 >= 0 && y0 < HH) ? 1.0f : 0.0f;
        const float vy1 = (y1 >= 0 && y1 < HH) ? 1.0f : 0.0f;

        const int cx0 = min(max(x0, 0), WW - 1);
        const int cx1 = min(max(x1, 0), WW - 1);
        const int cy0 = min(max(y0, 0), HH - 1);
        const int cy1 = min(max(y1, 0), HH - 1);

        w00[j] = wy0 * wx0 * (vy0 * vx0);
        w01[j] = wy0 * wx1 * (vy0 * vx1);
        w10[j] = wy1 * wx0 * (vy1 * vx0);
        w11[j] = wy1 * wx1 * (vy1 * vx1);

        // Derive the 4 corner offsets from one base + two deltas
        // (saves redundant mul/add index math).
        const int base = cy0 * WW + cx0;
        const int dx   = cx1 - cx0;          // 0 or 1
        const int dy   = (cy1 - cy0) * WW;   // 0 or WW
        o00[j] = base;
        o01[j] = base + dx;
        o10[j] = base + dy;
        o11[j] = base + dy + dx;
    }

    const float* imgb = img + (size_t)b * (CC * HH * WW);

    // gfx1250 global_prefetch_b8: warm the channel-1/2 gather lines while
    // channel 0's loads/FMAs are in flight (each batch image is ~3MB and
    // L2-resident; this pulls lines toward the WGP early).
    __builtin_prefetch(imgb + 1 * HH * WW + o00[0], 0, 1);
    __builtin_prefetch(imgb + 2 * HH * WW + o00[0], 0, 1);

    const size_t out_base = ((size_t)b * CC) * (size_t)(HH * WW)
                          + (size_t)y * WW + (size_t)xb;

#pragma unroll
    for (int c = 0; c < CC; ++c) {
        const float* __restrict__ p = imgb + c * (HH * WW);
        float4 v;
        float* vv = &v.x;
#pragma unroll
        for (int j = 0; j < 4; ++j) {
            vv[j] = w00[j] * p[o00[j]] + w01[j] * p[o01[j]]
                  + w10[j] * p[o10[j]] + w11[j] * p[o11[j]];
        }
        // 16B-aligned float4 store -> global_store_b128
        *reinterpret_cast<float4*>(out + out_base + (size_t)c * (HH * WW)) = v;
    }
}

extern "C" void kernel_launch(void* const* d_in, const int* in_sizes, int n_in,
                              void* d_out, int out_size, void* d_ws, size_t ws_size,
                              hipStream_t stream) {
    (void)in_sizes; (void)n_in; (void)out_size; (void)d_ws; (void)ws_size;
    const float* img   = (const float*)d_in[0];
    const float* rot   = (const float*)d_in[1];
    const float* trans = (const float*)d_in[2];
    const float* scale = (const float*)d_in[3];
    float* out = (float*)d_out;

    dim3 grid(HH / 2, BB);   // (256, 64): one block per (row-pair, batch)
    dim3 block(256);         // 8 wave32 waves
    warp_bilinear_kernel<<<grid, block, 0, stream>>>(img, rot, trans, scale, out);
}